// KernelConv_68315749810451
// MI455X (gfx1250) — compile-verified
//
#include <hip/hip_runtime.h>
#include <stdint.h>

// Per-pixel predicted-kernel 5x5 convolution (KPN/CARAFE style), fp32.
// Bandwidth-bound: ~63 MB min HBM traffic -> ~2.7 us roofline @ 23.3 TB/s.
// CDNA5 paths: global_load_async_to_lds_b32 (transposing stage, ASYNCcnt) +
// s_wait_asynccnt; conflict-free ds_load over channel-vector LDS layout;
// y-pair register blocking cuts LDS read traffic 1.67x (LDS bw is the
// limiting resource, not HBM); float2 accumulators to invite v_pk_fma_f32.

typedef __attribute__((ext_vector_type(2))) float v2f;

namespace {
constexpr int BS = 4, C = 64, H = 160, W = 160, K = 5;
constexpr int TX = 32, TY = 8;     // output pixel tile per workgroup
constexpr int CG = 32;             // channels per workgroup
constexpr int HR = TY + K - 1;     // 12 halo rows
constexpr int HC = TX + K - 1;     // 36 halo cols
constexpr int CPV = (CG + 4) / 4;  // 9 float4 -> 36-float channel pitch
                                   // (col stride 144B => bank step 36 mod 64:
                                   //  conflict-free across 16-lane groups)
constexpr int NTHR = 128;          // 32 tx * 4 y-pairs
}

__global__ __launch_bounds__(NTHR)
void kpn_conv_kernel(const float* __restrict__ fm,
                     const float* __restrict__ kern,
                     const int* __restrict__ dil_p,
                     float* __restrict__ out)
{
  const int dil = dil_p[0];   // uniform branch (scalar load)
  if (__builtin_expect(dil == 1, 1)) {
    __shared__ float4 lds4[HR * HC * CPV];          // 62,208 bytes
    float* ldsf = reinterpret_cast<float*>(lds4);

    int bid = blockIdx.x;
    const int nx = W / TX, ny = H / TY;             // 5, 20
    const int cg = bid & 1;  bid >>= 1;
    const int bx = bid % nx; bid /= nx;
    const int by = bid % ny; bid /= ny;
    const int b  = bid;                             // 0..3
    const int x0 = bx * TX, y0 = by * TY, c0 = cg * CG;

    const int tid = threadIdx.x;

    // ---- Stage fm halo (CG ch x 12 x 36) into LDS with async loads,
    //      transposing [c][y][x] (global) -> [y][x][c] (LDS). Global side:
    //      consecutive lanes hit consecutive x => fully coalesced b32.
    const uint64_t fm_base =
        (uint64_t)(uintptr_t)(fm + (size_t)(b * C + c0) * H * W);
    const uint32_t lds_base = (uint32_t)(uintptr_t)ldsf;

    constexpr int TOTAL = CG * HR * HC;             // 13824 = 108 iters * 128
    for (int idx = tid; idx < TOTAL; idx += NTHR) {
      const int cc = idx % HC;                      // halo col (innermost)
      const int t1 = idx / HC;
      const int r  = t1 % HR;                       // halo row
      const int c  = t1 / HR;                       // channel in group
      const int gy = y0 - (K / 2) + r;
      const int gx = x0 - (K / 2) + cc;
      const int lidx = (r * HC + cc) * (CPV * 4) + c;
      if ((unsigned)gy < (unsigned)H && (unsigned)gx < (unsigned)W) {
        const uint32_t laddr = lds_base + (uint32_t)(lidx * 4);
        const uint32_t goff  = (uint32_t)(((c * H + gy) * W + gx) * 4);
        asm volatile("global_load_async_to_lds_b32 %0, %1, %2"
                     :: "v"(laddr), "v"(goff), "s"(fm_base)
                     : "memory");
      } else {
        ldsf[lidx] = 0.0f;                          // zero-pad halo
      }
    }

    // ---- Per-pixel 5x5 weights for BOTH y's of the pair -> registers
    //      (flip over tap axis), loaded while async staging is in flight.
    const int tx  = tid & (TX - 1);                 // 0..31
    const int typ = tid >> 5;                       // y-pair 0..3
    const int gx  = x0 + tx;
    const int yA  = y0 + 2 * typ, yB = yA + 1;
    float wA[K * K], wB[K * K];
#pragma unroll
    for (int t = 0; t < K * K; ++t) {
      const size_t kp = (size_t)(b * K * K + (K * K - 1 - t)) * H;
      wA[t] = kern[(kp + yA) * W + gx];
      wB[t] = kern[(kp + yB) * W + gx];
    }

    asm volatile("s_wait_asynccnt 0x0" ::: "memory");
    __syncthreads();

    // ---- Compute: 8 channel-quads; y-pair shares the 6 halo rows, so the
    //      pair needs 6x5=30 b128 LDS reads per quad instead of 2x25=50.
    const size_t plane = (size_t)H * W;
#pragma unroll 1
    for (int cb4 = 0; cb4 < CG / 4; ++cb4) {
      v2f aA0 = {0.f, 0.f}, aA1 = {0.f, 0.f};      // pixel (gx, yA), ch lo/hi
      v2f aB0 = {0.f, 0.f}, aB1 = {0.f, 0.f};      // pixel (gx, yB), ch lo/hi
#pragma unroll
      for (int i2 = 0; i2 < K + 1; ++i2) {         // 6 shared halo rows
        const int rbase = (2 * typ + i2) * HC;
#pragma unroll
        for (int j = 0; j < K; ++j) {
          const float4 v = lds4[(rbase + (tx + j)) * CPV + cb4];
          const v2f vlo = {v.x, v.y};
          const v2f vhi = {v.z, v.w};
          if (i2 < K) {                            // tap row i=i2 for yA
            const float w = wA[i2 * K + j];
            const v2f w2 = {w, w};
            aA0 += w2 * vlo;
            aA1 += w2 * vhi;
          }
          if (i2 > 0) {                            // tap row i=i2-1 for yB
            const float w = wB[(i2 - 1) * K + j];
            const v2f w2 = {w, w};
            aB0 += w2 * vlo;
            aB1 += w2 * vhi;
          }
        }
      }
      size_t obA = ((size_t)(b * C + c0 + cb4 * 4) * H + yA) * W + gx;
      size_t obB = obA + W;                         // yB = yA + 1
      out[obA]             = aA0.x;
      out[obA + plane]     = aA0.y;
      out[obA + 2 * plane] = aA1.x;
      out[obA + 3 * plane] = aA1.y;
      out[obB]             = aB0.x;
      out[obB + plane]     = aB0.y;
      out[obB + 2 * plane] = aB1.x;
      out[obB + 3 * plane] = aB1.y;
    }
  } else {
    // ---- General-dilation fallback (harness always passes dilation==1).
    const long total = (long)BS * C * H * W;
    for (long idx = (long)blockIdx.x * blockDim.x + threadIdx.x; idx < total;
         idx += (long)gridDim.x * blockDim.x) {
      int x = (int)(idx % W); long t = idx / W;
      int y = (int)(t % H);   t /= H;
      int c = (int)(t % C);   int b = (int)(t / C);
      float acc = 0.f;
      for (int i = 0; i < K; ++i) {
        for (int j = 0; j < K; ++j) {
          const int yy = y + (i - K / 2) * dil;
          const int xx = x + (j - K / 2) * dil;
          float v = 0.f;
          if ((unsigned)yy < (unsigned)H && (unsigned)xx < (unsigned)W)
            v = fm[((size_t)(b * C + c) * H + yy) * W + xx];
          const float wv =
              kern[((size_t)(b * K * K + (K * K - 1 - (i * K + j))) * H + y) * W + x];
          acc = fmaf(wv, v, acc);
        }
      }
      out[idx] = acc;
    }
  }
}

extern "C" void kernel_launch(void* const* d_in, const int* in_sizes, int n_in,
                              void* d_out, int out_size, void* d_ws, size_t ws_size,
                              hipStream_t stream) {
  (void)in_sizes; (void)n_in; (void)out_size; (void)d_ws; (void)ws_size;
  const float* fm   = (const float*)d_in[0];
  const float* kern = (const float*)d_in[1];
  const int*   dil  = (const int*)d_in[2];
  float*       out  = (float*)d_out;

  const int grid = BS * (H / TY) * (W / TX) * (C / CG);  // 4*20*5*2 = 800
  kpn_conv_kernel<<<grid, NTHR, 0, stream>>>(fm, kern, dil, out);
}